// Diffusion_9715216023975
// MI455X (gfx1250) — compile-verified
//
#include <hip/hip_runtime.h>
#include <hip/hip_bf16.h>

typedef float v2f __attribute__((ext_vector_type(2)));
typedef float v8f __attribute__((ext_vector_type(8)));

#define TPB   256
#define VEC   4
#define ITERS 4   // elements per thread = 16 -> 4096 per block

// Deterministic counter-based RNG (PCG hash), keyed on flat edge index.
__device__ __forceinline__ unsigned pcg_hash(unsigned v) {
    unsigned s = v * 747796405u + 2891336453u;
    unsigned w = ((s >> ((s >> 28) + 4u)) ^ s) * 277803737u;
    return (w >> 22) ^ w;
}

__global__ void Diffusion_init_out(float* out) { out[0] = 0.0f; }

__global__ __launch_bounds__(TPB)
void Diffusion_bce_kernel(const int* __restrict__ adj,
                          const int* __restrict__ t_arr,
                          const float* __restrict__ q_approx,
                          const float* __restrict__ Qt,
                          float* __restrict__ out,
                          int total, int perB, int T, float scale) {
    const int block_base = blockIdx.x * (TPB * VEC * ITERS);
    // 4096-element chunks never straddle a batch (perB = N*N is a multiple of 4096)
    const int b   = block_base / perB;
    const int tb  = t_arr[b];                       // uniform per block -> scalar load
    const int tm1 = (tb == 0) ? (T - 1) : (tb - 1); // JAX Qt[t-1] wraps at t==0
    // Qt[t] = [[nf,f],[f,nf]] flattened at 4*t
    const float nf_t = Qt[4 * tb + 0],  f_t = Qt[4 * tb + 1];
    const float nf_p = Qt[4 * tm1 + 0], f_p = Qt[4 * tm1 + 1];
    const float nf_1 = Qt[0],           f_1 = Qt[1];

    float acc = 0.0f;
#pragma unroll
    for (int it = 0; it < ITERS; ++it) {
        const int idx = block_base + it * (TPB * VEC) + (int)threadIdx.x * VEC;
        if (idx < total) {
            const int4   a4 = *reinterpret_cast<const int4*>(adj + idx);
            const float4 q4 = *reinterpret_cast<const float4*>(q_approx + idx);
            // streaming prefetch hint one block-chunk ahead (speculative; OOB dropped)
            __builtin_prefetch(q_approx + idx + TPB * VEC * ITERS, 0, 0);
            __builtin_prefetch(adj      + idx + TPB * VEC * ITERS, 0, 0);
#pragma unroll
            for (int e = 0; e < VEC; ++e) {
                const int   a = (&a4.x)[e];
                const float q = (&q4.x)[e];
                // forward diffusion: row Qt[tb][a] = [P0, 1-P0]
                const float P0 = a ? f_t : nf_t;
                const unsigned h = pcg_hash((unsigned)(idx + e) ^ 0x2545F491u); // key 42 mix
                const float u = (float)(h >> 8) * (1.0f / 16777216.0f);
                const int   x = (u >= P0) ? 1 : 0;   // inverse-CDF categorical
                // posterior q(x_{t-1}=0 | x_t, x_0)
                const float lik0   = x ? f_1 : nf_1;     // Qt[0][x][0]
                const float prior0 = a ? f_p : nf_p;     // Qt[tm1][a][0]
                const float ev     = x ? (1.0f - P0) : P0;
                const float qt     = lik0 * prior0 / ev;
                // BCE with torch-style log clamping
                const float logp   = fmaxf(__logf(q), -100.0f);
                const float log1mp = fmaxf(__logf(1.0f - q), -100.0f);
                acc += qt * logp + (1.0f - qt) * log1mp;
            }
        }
    }

    // ---- wave32 reduction via V_WMMA_F32_16X16X4_F32: D = A(acc) x B(ones) ----
    // A layout (16x4 f32): lanes 0-15 vgpr0 -> K=0, lanes 16-31 vgpr0 -> K=2; vgpr1 zeroed.
    // => D[m][n] = acc_m + acc_{m+16}. Lane sums its 8 D vgprs (S_0..7 or S_8..15),
    // then one xor-16 exchange yields the full wave total in every lane.
    float wtot;
#if defined(__gfx1250__) && __has_builtin(__builtin_amdgcn_wmma_f32_16x16x4_f32)
    {
        v2f afrag; afrag[0] = acc;  afrag[1] = 0.0f;
        v2f bfrag; bfrag[0] = 1.0f; bfrag[1] = 1.0f;
        v8f c = {};
        v8f d = __builtin_amdgcn_wmma_f32_16x16x4_f32(
            /*neg_a=*/false, afrag, /*neg_b=*/false, bfrag,
            /*c_mod=*/(short)0, c, /*reuse_a=*/false, /*reuse_b=*/false);
        float s = d[0] + d[1] + d[2] + d[3] + d[4] + d[5] + d[6] + d[7];
        wtot = s + __shfl_xor(s, 16, 32);
    }
#else
    {
        float s = acc;
        for (int off = 16; off; off >>= 1) s += __shfl_xor(s, off, 32);
        wtot = s;
    }
#endif

    __shared__ float lsum;
    if (threadIdx.x == 0) lsum = 0.0f;
    __syncthreads();
    if ((threadIdx.x & 31) == 0) atomicAdd(&lsum, wtot);   // ds_add_f32
    __syncthreads();
    if (threadIdx.x == 0) {
#if defined(__HIP_DEVICE_COMPILE__)
        unsafeAtomicAdd(out, lsum * scale);                // global_atomic_add_f32
#else
        atomicAdd(out, lsum * scale);
#endif
    }
}

extern "C" void kernel_launch(void* const* d_in, const int* in_sizes, int n_in,
                              void* d_out, int out_size, void* d_ws, size_t ws_size,
                              hipStream_t stream) {
    const int*   adj = (const int*)d_in[0];    // adj_start [B,N,N] int32
    const int*   t   = (const int*)d_in[1];    // t [B] int32
    const float* q   = (const float*)d_in[2];  // q_approx [B*N*N] f32
    const float* Qt  = (const float*)d_in[3];  // Qt [T,2,2] f32
    float* out = (float*)d_out;

    const int total = in_sizes[0];             // B*N*N = 16,777,216
    const int B     = in_sizes[1];
    const int T     = in_sizes[3] / 4;
    const int perB  = total / B;               // N*N
    const int epb   = TPB * VEC * ITERS;       // 4096
    const int blocks = (total + epb - 1) / epb;
    const float scale = -1.0f / (float)total;  // negative mean

    hipLaunchKernelGGL(Diffusion_init_out, dim3(1), dim3(1), 0, stream, out);
    hipLaunchKernelGGL(Diffusion_bce_kernel, dim3(blocks), dim3(TPB), 0, stream,
                       adj, t, q, Qt, out, total, perB, T, scale);
}